// ProposalLayer_35708358099154
// MI455X (gfx1250) — compile-verified
//
#include <hip/hip_runtime.h>
#include <math.h>

// ---------------------------------------------------------------------------
// Faster-RCNN ProposalLayer for MI455X (gfx1250, wave32)
//   B=4, A=9, H=38, W=63 -> N = 21546 anchors/image
//   exact top-6000 by score (sorted), greedy NMS @0.7, first 300 kept
//   -> [b, x1, y1, x2, y2]
//
// 4 launches total:
//   1) decode + 8192-tile LDS bitonic sort       (B*4 blocks, 1024 thr)
//   2) bitonic half-clean merge 4->2 tiles       (B*2 blocks, 1024 thr)
//   3) merge 2->1 tiles + gather top-6000 boxes  (B   blocks, 1024 thr)
//   4) blocked greedy NMS (32-candidate groups,  (B   blocks, 256 thr)
//      2 barriers/group, async-LDS double buffer)
// ---------------------------------------------------------------------------

#define NUM_A      9
#define FH         38
#define FW         63
#define NTOT       (NUM_A * FH * FW)   // 21546
#define NSORT      32768               // 4 tiles of 8192
#define TILE       8192
#define TPB        1024
#define PRE_NMS_K  6000
#define POST_NMS_K 300
#define NMS_TH     0.7f
#define CHUNK      512                 // NMS streaming chunk (boxes)

// Classic 9 Faster-RCNN anchors (base 16, ratios .5/1/2, scales 8/16/32)
__constant__ float ANCH[NUM_A][4] = {
    {-84.f,  -40.f,  99.f,  55.f},
    {-176.f, -88.f,  191.f, 103.f},
    {-360.f, -184.f, 375.f, 199.f},
    {-56.f,  -56.f,  71.f,  71.f},
    {-120.f, -120.f, 135.f, 135.f},
    {-248.f, -248.f, 263.f, 263.f},
    {-36.f,  -80.f,  51.f,  95.f},
    {-80.f,  -168.f, 95.f,  183.f},
    {-168.f, -344.f, 183.f, 359.f}
};

// ---------------------------------------------------------------------------
// CDNA5 async global->LDS helpers (ASYNCcnt path, wave32).
// Generic pointer to __shared__: low 32 bits are the LDS byte offset
// (ISA: LDS aperture lives in ADDR[63:32]).
// ---------------------------------------------------------------------------
__device__ __forceinline__ void async_copy_b128_to_lds(void* lds_ptr, const void* gptr) {
    unsigned lds_off = (unsigned)(size_t)lds_ptr;
    unsigned long long ga = (unsigned long long)(size_t)gptr;
    asm volatile("global_load_async_to_lds_b128 %0, %1, off"
                 :: "v"(lds_off), "v"(ga)
                 : "memory");
}
__device__ __forceinline__ void wait_async0() {
    asm volatile("s_wait_asynccnt 0x0" ::: "memory");
}

// total order: key descending, index ascending on ties (lax.top_k stability)
__device__ __forceinline__ bool kv_gt(float ka, int va, float kb, int vb) {
    return (ka > kb) || ((ka == kb) && (va < vb));
}

// IoU > thresh predicate, reference math (division, +1 widths)
__device__ __forceinline__ bool iou_over(const float4& a, float aarea,
                                         const float4& b, float barea) {
    float xx1 = fmaxf(a.x, b.x);
    float yy1 = fmaxf(a.y, b.y);
    float xx2 = fminf(a.z, b.z);
    float yy2 = fminf(a.w, b.w);
    float iw  = fmaxf(xx2 - xx1 + 1.0f, 0.0f);
    float ih  = fmaxf(yy2 - yy1 + 1.0f, 0.0f);
    float inter = iw * ih;
    float iou = inter / (aarea + barea - inter);
    return iou > NMS_TH;
}
__device__ __forceinline__ float box_area(const float4& a) {
    return (a.z - a.x + 1.0f) * (a.w - a.y + 1.0f);
}

// ---------------------------------------------------------------------------
// Kernel 1: decode 8192 anchors + full LDS bitonic sort of the tile
// ---------------------------------------------------------------------------
__global__ __launch_bounds__(TPB)
void decode_sort_kernel(const float* __restrict__ cls,
                        const float* __restrict__ bbox,
                        const float* __restrict__ iminfo,
                        float* __restrict__ props,   // (B, NTOT, 4)
                        float* __restrict__ keys,    // (B, NSORT)
                        int*   __restrict__ vals) {  // (B, NSORT)
    __shared__ float sk[TILE];   // 32 KB
    __shared__ int   sv[TILE];   // 32 KB
    const int tid      = threadIdx.x;
    const int tile     = blockIdx.x & 3;   // 4 tiles per batch
    const int b        = blockIdx.x >> 2;
    const int tileBase = tile * TILE;

    const float maxx = iminfo[b * 3 + 1] - 1.0f;
    const float maxy = iminfo[b * 3 + 0] - 1.0f;
    const size_t plane = (size_t)FH * FW;

    // ---- decode 8 elements per thread ----
    for (int q = 0; q < TILE / TPB; ++q) {
        int local = q * TPB + tid;
        int i = tileBase + local;
        float key = -INFINITY;
        if (i < NTOT) {
            int a = i % NUM_A;
            int s = i / NUM_A;
            int w = s % FW;
            int h = s / FW;

            key = cls[((size_t)b * (2 * NUM_A) + NUM_A + a) * plane + (size_t)h * FW + w];

            size_t dbase = ((size_t)b * (4 * NUM_A) + 4 * a) * plane + (size_t)h * FW + w;
            float dx = bbox[dbase];
            float dy = bbox[dbase + plane];
            float dw = bbox[dbase + 2 * plane];
            float dh = bbox[dbase + 3 * plane];

            float ax1 = ANCH[a][0] + (float)(w * 16);
            float ay1 = ANCH[a][1] + (float)(h * 16);
            float ax2 = ANCH[a][2] + (float)(w * 16);
            float ay2 = ANCH[a][3] + (float)(h * 16);

            float ww = ax2 - ax1 + 1.0f;
            float hh = ay2 - ay1 + 1.0f;
            float cx = ax1 + 0.5f * ww;
            float cy = ay1 + 0.5f * hh;

            float pcx = dx * ww + cx;
            float pcy = dy * hh + cy;
            float pw  = expf(dw) * ww;
            float ph  = expf(dh) * hh;

            float px1 = fminf(fmaxf(pcx - 0.5f * pw, 0.0f), maxx);
            float py1 = fminf(fmaxf(pcy - 0.5f * ph, 0.0f), maxy);
            float px2 = fminf(fmaxf(pcx + 0.5f * pw, 0.0f), maxx);
            float py2 = fminf(fmaxf(pcy + 0.5f * ph, 0.0f), maxy);

            ((float4*)props)[(size_t)b * NTOT + i] = make_float4(px1, py1, px2, py2);
        }
        sk[local] = key;
        sv[local] = i;        // within-batch index (also tie-break key)
    }
    __syncthreads();

    // ---- full bitonic sort of the tile, descending ----
    for (int k = 2; k <= TILE; k <<= 1) {
        for (int j = k >> 1; j >= 1; j >>= 1) {
            for (int tt = tid; tt < TILE / 2; tt += TPB) {
                int i = ((tt & ~(j - 1)) << 1) | (tt & (j - 1));
                int p = i + j;
                bool desc = ((i & k) == 0);
                float ka = sk[i], kb = sk[p];
                int   va = sv[i], vb = sv[p];
                bool sw = desc ? kv_gt(kb, vb, ka, va) : kv_gt(ka, va, kb, vb);
                if (sw) { sk[i] = kb; sk[p] = ka; sv[i] = vb; sv[p] = va; }
            }
            __syncthreads();
        }
    }

    // ---- write sorted tile back ----
    size_t base = (size_t)b * NSORT + tileBase;
    for (int q = 0; q < TILE / TPB; ++q) {
        int local = q * TPB + tid;
        keys[base + local] = sk[local];
        vals[base + local] = sv[local];
    }
}

// ---------------------------------------------------------------------------
// Kernel 2/3: bitonic half-cleaner merge of two descending 8192 runs.
// max(a[i], b[n-1-i]) is exactly the top-n multiset and is bitonic; 13
// same-direction stages finish the sort. A-tile staged via async->LDS.
// doGather: gather top-6000 boxes into contiguous sprops instead of
// writing keys/vals back.
// ---------------------------------------------------------------------------
__global__ __launch_bounds__(TPB)
void merge_kernel(float* __restrict__ keys, int* __restrict__ vals,
                  const float* __restrict__ props, float* __restrict__ sprops,
                  int strideTiles, int nPairs, int doGather) {
    __shared__ float sk[TILE];
    __shared__ int   sv[TILE];
    const int tid = threadIdx.x;
    const int b   = blockIdx.x / nPairs;
    const int q   = blockIdx.x % nPairs;

    const size_t batchBase = (size_t)b * NSORT;
    const size_t aOff = batchBase + (size_t)q * 2 * strideTiles * TILE;
    const size_t bOff = aOff + (size_t)strideTiles * TILE;

    // ---- async stage a-tile (keys + vals) into LDS: b128 = 4 dwords ----
    for (int e = tid; e < TILE / 4; e += TPB) {
        async_copy_b128_to_lds(&sk[e * 4], keys + aOff + (size_t)e * 4);
        async_copy_b128_to_lds(&sv[e * 4], vals + aOff + (size_t)e * 4);
    }
    wait_async0();
    __syncthreads();

    // ---- half-cleaner vs reversed b-tile: keep the larger of each pair ----
    for (int e = tid; e < TILE; e += TPB) {
        size_t r = bOff + (size_t)(TILE - 1 - e);
        if (e + TPB < TILE)   // gfx1250 global_prefetch_b8 ahead on the reversed stream
            __builtin_prefetch(&keys[bOff + (size_t)(TILE - 1 - (e + TPB))], 0, 0);
        float kb = keys[r];
        int   vb = vals[r];
        if (kv_gt(kb, vb, sk[e], sv[e])) { sk[e] = kb; sv[e] = vb; }
    }
    __syncthreads();

    // ---- descending bitonic merge of the (bitonic) max half ----
    for (int j = TILE / 2; j >= 1; j >>= 1) {
        for (int tt = tid; tt < TILE / 2; tt += TPB) {
            int i = ((tt & ~(j - 1)) << 1) | (tt & (j - 1));
            int p = i + j;
            float ka = sk[i], kb = sk[p];
            int   va = sv[i], vb = sv[p];
            if (kv_gt(kb, vb, ka, va)) { sk[i] = kb; sk[p] = ka; sv[i] = vb; sv[p] = va; }
        }
        __syncthreads();
    }

    if (doGather) {
        // final round: emit contiguous top-6000 boxes in rank order
        for (int r = tid; r < PRE_NMS_K; r += TPB) {
            int idx = sv[r];   // finite keys (>=6000 real entries) => idx < NTOT
            ((float4*)sprops)[(size_t)b * PRE_NMS_K + r] =
                ((const float4*)props)[(size_t)b * NTOT + idx];
        }
    } else {
        for (int e = tid; e < TILE; e += TPB) {
            keys[aOff + e] = sk[e];
            vals[aOff + e] = sv[e];
        }
    }
}

// ---------------------------------------------------------------------------
// Kernel 4: blocked greedy NMS — one workgroup (8 wave32) per batch.
// Candidates handled in groups of 32:
//   phase 1 (parallel, 256 thr): group-vs-kept suppression bits +
//            32x32 intra-group IoU bit-matrix (rows hold bits l>m only)
//   phase 2 (uniform sweep, no barriers inside): exact greedy recurrence
//            over bitmasks; wave-0 lanes append kept boxes via popc-prefix
// Accumulators are ping-ponged by group parity so zeroing overlaps reads:
// only 2 block barriers per 32 candidates (vs 2 per candidate before).
// Sorted boxes stream through double-buffered async-LDS chunks.
// ---------------------------------------------------------------------------
__global__ __launch_bounds__(256)
void nms_kernel(const float* __restrict__ sprops, float* __restrict__ out) {
    const int b   = blockIdx.x;
    const int tid = threadIdx.x;
    __shared__ float4   sbuf[2][CHUNK];      // streamed candidates (16 KB)
    __shared__ float4   kbox[POST_NMS_K];    // kept boxes
    __shared__ float    karea[POST_NMS_K];   // kept areas
    __shared__ unsigned s_supp[2];           // per-parity: kept-suppression bits
    __shared__ unsigned gmat[2][32];         // per-parity: intra-group kill rows
    __shared__ int      s_nk;

    // init all 300 output rows for this batch: [b, 0,0,0,0]
    for (int r = tid; r < POST_NMS_K; r += 256) {
        float* row = out + ((size_t)b * POST_NMS_K + r) * 5;
        row[0] = (float)b;
        row[1] = 0.0f; row[2] = 0.0f; row[3] = 0.0f; row[4] = 0.0f;
    }
    if (tid < 32) { gmat[0][tid] = 0u; gmat[1][tid] = 0u; }
    if (tid == 0) { s_supp[0] = 0u; s_supp[1] = 0u; }

    const float4* boxes = (const float4*)sprops + (size_t)b * PRE_NMS_K;

    // async prefetch chunk 0 into LDS
    for (int t = tid; t < CHUNK; t += 256) {
        int gi = (t < PRE_NMS_K) ? t : (PRE_NMS_K - 1);
        async_copy_b128_to_lds(&sbuf[0][t], boxes + gi);
    }
    wait_async0();
    __syncthreads();

    int nk = 0;
    int gg = 0;   // global group counter (parity source)
    const int nchunks = (PRE_NMS_K + CHUNK - 1) / CHUNK;
    for (int c = 0; c < nchunks && nk < POST_NMS_K; ++c) {
        // kick off async load of next chunk while we process this one
        if (c + 1 < nchunks) {
            int nb = (c + 1) * CHUNK;
            for (int t = tid; t < CHUNK; t += 256) {
                int gi = nb + t;
                if (gi >= PRE_NMS_K) gi = PRE_NMS_K - 1;
                async_copy_b128_to_lds(&sbuf[(c + 1) & 1][t], boxes + gi);
            }
        }

        const float4* cur = sbuf[c & 1];
        int climit = PRE_NMS_K - c * CHUNK;
        if (climit > CHUNK) climit = CHUNK;

        for (int g0 = 0; g0 < climit && nk < POST_NMS_K; g0 += 32, ++gg) {
            const int gsz = (climit - g0 < 32) ? (climit - g0) : 32;
            const int par = gg & 1;

            // ---------------- phase 1 (all threads, no internal barriers) ---
            // (a) zero next-parity accumulators (disjoint from this group's)
            if (tid < 32) gmat[par ^ 1][tid] = 0u;
            if (tid == 32) s_supp[par ^ 1] = 0u;

            // (b) group candidates vs kept list: accumulate locally, 1 atomic
            {
                unsigned localbits = 0u;
                int npair = nk << 5;                     // 32 * nk
                for (int p = tid; p < npair; p += 256) {
                    int l  = p & 31;
                    int kq = p >> 5;
                    if (l < gsz) {
                        float4 cb = cur[g0 + l];
                        if (iou_over(cb, box_area(cb), kbox[kq], karea[kq]))
                            localbits |= (1u << l);
                    }
                }
                if (localbits) atomicOr(&s_supp[par], localbits);
            }

            // (c) intra-group 32x32 IoU bit-matrix; row m gets bits l > m
            for (int p = tid; p < 32 * 32; p += 256) {
                int l = p & 31;
                int m = p >> 5;
                if (l > m && l < gsz && m < gsz) {
                    float4 cl = cur[g0 + l];
                    float4 cm = cur[g0 + m];
                    if (iou_over(cl, box_area(cl), cm, box_area(cm)))
                        atomicOr(&gmat[par][m], 1u << l);
                }
            }
            __syncthreads();

            // ---------------- phase 2: uniform greedy sweep -----------------
            unsigned invalid  = (gsz == 32) ? 0u : ~((1u << gsz) - 1u);
            unsigned killed   = s_supp[par] | invalid;
            for (int m = 0; m < gsz; ++m)
                if (!((killed >> m) & 1u))
                    killed |= gmat[par][m];        // rows only hold bits l > m
            unsigned keepbits = ~killed & ~invalid;

            // wave-0 lanes append kept candidates in rank order
            if (tid < 32) {
                int l = tid;
                if ((keepbits >> l) & 1u) {
                    int rank = __popc(keepbits & ((1u << l) - 1u));
                    int slot = nk + rank;
                    if (slot < POST_NMS_K) {
                        float4 bx = cur[g0 + l];
                        kbox[slot]  = bx;
                        karea[slot] = box_area(bx);
                        float* row = out + ((size_t)b * POST_NMS_K + slot) * 5;
                        row[1] = bx.x; row[2] = bx.y; row[3] = bx.z; row[4] = bx.w;
                    }
                }
            }
            if (tid == 0) {
                int add = __popc(keepbits);
                int nn  = nk + add;
                s_nk = (nn > POST_NMS_K) ? POST_NMS_K : nn;
            }
            __syncthreads();
            nk = s_nk;
        }

        wait_async0();     // next buffer resident before we flip
        __syncthreads();
    }
}

// ---------------------------------------------------------------------------
// launcher — 4 kernels total
// ---------------------------------------------------------------------------
extern "C" void kernel_launch(void* const* d_in, const int* in_sizes, int n_in,
                              void* d_out, int out_size, void* d_ws, size_t ws_size,
                              hipStream_t stream) {
    const float* cls    = (const float*)d_in[0];   // (B, 2A, H, W)
    const float* bbox   = (const float*)d_in[1];   // (B, 4A, H, W)
    const float* iminfo = (const float*)d_in[2];   // (B, 3)
    float* out = (float*)d_out;                    // (B, 300, 5)

    const int B = in_sizes[2] / 3;                 // 4

    // workspace layout (16B aligned slabs)
    char* ws = (char*)d_ws;
    size_t props_bytes  = (size_t)B * NTOT * 4 * sizeof(float);       // 1.38 MB
    size_t sprops_bytes = (size_t)B * PRE_NMS_K * 4 * sizeof(float);  // 384 KB
    size_t keys_bytes   = (size_t)B * NSORT * sizeof(float);          // 512 KB
    float* props  = (float*)ws;
    float* sprops = (float*)(ws + props_bytes);
    float* keys   = (float*)(ws + props_bytes + sprops_bytes);
    int*   vals   = (int*)  (ws + props_bytes + sprops_bytes + keys_bytes);

    // 1) decode + per-tile LDS sort (4 tiles of 8192 per batch)
    decode_sort_kernel<<<B * 4, TPB, 0, stream>>>(cls, bbox, iminfo, props, keys, vals);

    // 2) exact top-k halving merges: 4 -> 2 tiles
    merge_kernel<<<B * 2, TPB, 0, stream>>>(keys, vals, props, sprops,
                                            /*stride*/1, /*nPairs*/2, /*gather*/0);

    // 3) 2 -> 1 tile (exact sorted top-8192) + gather top-6000 boxes
    merge_kernel<<<B * 1, TPB, 0, stream>>>(keys, vals, props, sprops,
                                            /*stride*/2, /*nPairs*/1, /*gather*/1);

    // 4) blocked greedy NMS + output, one WGP workgroup per batch
    nms_kernel<<<B, 256, 0, stream>>>(sprops, out);
}